// DFMBPSROIAlign_55946243997743
// MI455X (gfx1250) — compile-verified
//
#include <hip/hip_runtime.h>
#include <stdint.h>

// Problem constants (from reference)
#define CCH   10
#define PHH   7
#define PWW   7
#define HH    34
#define WW    34
#define NBINS (PHH * PWW)                   // 49
#define CH_SZ (HH * WW)                     // 1156 pixels per (c,bin) slice
#define FT_CH_STRIDE (PHH * PWW * HH * WW)  // 56644 floats between channels
#define PIX   12                            // 10 channels padded to 12 -> 48B/pixel (16B aligned)
#define LDS_FLOATS (CH_SZ * PIX)            // 13872 floats = 55488 bytes
#define ROI_TILE 256
#define TAIL (CH_SZ - 4 * ROI_TILE)         // 132

typedef __attribute__((ext_vector_type(2))) float v2f;

// Load 10 interleaved channel values for one pixel (b128 + b128 + b64 from LDS)
// and expose them as five packed-f32 pairs (register re-labeling only).
__device__ __forceinline__ void load10v(const float* __restrict__ p, v2f v[5]) {
    const float4 a = *(const float4*)(p + 0);
    const float4 b = *(const float4*)(p + 4);
    const float2 c = *(const float2*)(p + 8);
    v[0].x = a.x; v[0].y = a.y;
    v[1].x = a.z; v[1].y = a.w;
    v[2].x = b.x; v[2].y = b.y;
    v[3].x = b.z; v[3].y = b.w;
    v[4].x = c.x; v[4].y = c.y;
}

// GVS form: uniform 64-bit base in SGPRs + per-lane 32-bit byte offset.
__device__ __forceinline__ void async_b32_gvs(uint32_t lds_off, uint32_t voff,
                                              uint64_t sbase) {
    asm volatile("global_load_async_to_lds_b32 %0, %1, %2"
                 :: "v"(lds_off), "v"(voff), "s"(sbase) : "memory");
}

__global__ __launch_bounds__(ROI_TILE)
void dfmb_psroi_align_kernel(const float* __restrict__ ft,
                             const float* __restrict__ rois,
                             float* __restrict__ out,
                             int N) {
    __shared__ __align__(16) float s_ft[LDS_FLOATS];

    const int bin = blockIdx.y;          // 0..48, bin = ph*PW + pw
    const int tid = threadIdx.x;

    // ---------------------------------------------------------------
    // Async global->LDS scatter: channel-interleave the per-bin slice
    // ft[:, ph, pw, :, :] into pixel-major layout s_ft[pixel*12 + c].
    // Uniform global base in SGPRs (GVS), 32-bit lane offsets; 4 rounds
    // + tail, fully unrolled; <= 50 outstanding async b32 per lane.
    // ---------------------------------------------------------------
    {
        const uint32_t lds0 = (uint32_t)(uintptr_t)(&s_ft[0]);
        #pragma unroll
        for (int c = 0; c < CCH; ++c) {
            const uint64_t gbase = (uint64_t)(uintptr_t)
                (ft + (size_t)bin * CH_SZ + (size_t)c * FT_CH_STRIDE);
            #pragma unroll
            for (int k = 0; k < 4; ++k) {
                const uint32_t p = (uint32_t)tid + (uint32_t)(k * ROI_TILE);
                async_b32_gvs(lds0 + (p * PIX + (uint32_t)c) * 4u, p * 4u, gbase);
            }
        }
        if (tid < TAIL) {
            const uint32_t p = (uint32_t)tid + 4u * ROI_TILE;
            #pragma unroll
            for (int c = 0; c < CCH; ++c) {
                const uint64_t gbase = (uint64_t)(uintptr_t)
                    (ft + (size_t)bin * CH_SZ + (size_t)c * FT_CH_STRIDE);
                async_b32_gvs(lds0 + (p * PIX + (uint32_t)c) * 4u, p * 4u, gbase);
            }
        }
    }

    // ---------------------------------------------------------------
    // Per-ROI geometry (independent of LDS) — hides async latency.
    // ---------------------------------------------------------------
    const int n  = blockIdx.x * ROI_TILE + tid;
    const int nc = (n < N) ? n : (N - 1);   // clamp for safe loads; store is guarded

    const float rsw = rois[nc * 5 + 1] / 8.0f;
    const float rsh = rois[nc * 5 + 2] / 8.0f;
    const float rew = rois[nc * 5 + 3] / 8.0f;
    const float reh = rois[nc * 5 + 4] / 8.0f;

    const float dh  = reh - rsh;
    const float dwf = rew - rsw;
    const float roi_h = (dh  > 0.1f) ? dh  : 0.1f;
    const float roi_w = (dwf > 0.1f) ? dwf : 0.1f;

    const float bsh = roi_h / 7.0f;
    const float bsw = roi_w / 7.0f;
    const float sub_h = bsh * 0.25f;
    const float sub_w = bsw * 0.25f;

    const int ph = bin / PWW;
    const int pw = bin - ph * PWW;

    const float hstart = floorf(rsh + (float)ph * bsh);
    const float wstart = floorf(rsw + (float)pw * bsw);

    // -----------------------------------------------------------------
    // Per-axis geometry, pre-masked by the per-axis keep flags:
    //   ax0/ax1 = kw ? (1-dx)/dx : 0 ;  ay0/ay1 = kh ? (1-dy)/dy : 0
    // so every cross product c12/c21/c22 vanishes when !keep, branchless.
    // Only bad11 (mixes axes) needs a per-combo select.
    // The sample count is separable: cnt = (sum kw) * (sum kh) — exact in
    // fp32 (small integers) — computed here, before the barrier.
    // -----------------------------------------------------------------
    float ax0[4], ax1[4], ay0[4], ay1[4];
    int   cx1[4], cx2[4];        // column offsets (floats): x*PIX
    int   ry1[4], ry2[4];        // row offsets (floats):    y*WW*PIX
    bool  bx[4], by[4];
    float kwsum = 0.0f, khsum = 0.0f;

    #pragma unroll
    for (int i = 0; i < 4; ++i) {
        const float w = wstart + ((float)i + 0.5f) * sub_w;
        const bool kw = (w > -1.0f) && (w < (float)WW);
        int x1 = (int)floorf(w);
        int x2 = (int)ceilf(w);
        const bool x1v = (x1 >= 0) && (x1 < WW);
        const bool x2v = (x2 >= 0) && (x2 < WW);
        x1 = min(max(x1, 0), WW - 1);
        x2 = min(max(x2, 0), WW - 1);
        const float dx = w - (float)x1;      // uses clipped x1 (matches reference)
        ax0[i] = kw ? (1.0f - dx) : 0.0f;
        ax1[i] = kw ? dx : 0.0f;
        kwsum += kw ? 1.0f : 0.0f;
        bx[i]  = (!x1v) || (!x2v);
        cx1[i] = x1 * PIX;
        cx2[i] = x2 * PIX;

        const float h = hstart + ((float)i + 0.5f) * sub_h;
        const bool kh = (h > -1.0f) && (h < (float)HH);
        int y1 = (int)floorf(h);
        int y2 = (int)ceilf(h);
        const bool y1v = (y1 >= 0) && (y1 < HH);
        const bool y2v = (y2 >= 0) && (y2 < HH);
        y1 = min(max(y1, 0), HH - 1);
        y2 = min(max(y2, 0), HH - 1);
        const float dy = h - (float)y1;      // uses clipped y1 (matches reference)
        ay0[i] = kh ? (1.0f - dy) : 0.0f;
        ay1[i] = kh ? dy : 0.0f;
        khsum += kh ? 1.0f : 0.0f;
        by[i]  = y1v || y2v;
        ry1[i] = y1 * (WW * PIX);
        ry2[i] = y2 * (WW * PIX);
    }

    const float cnt = kwsum * khsum;   // == number of kept subsamples (exact)

    // Wait for the staged slice, then sync the workgroup.
    asm volatile("s_wait_asynccnt 0" ::: "memory");
    __syncthreads();

    v2f sum2[5];
    #pragma unroll
    for (int j = 0; j < 5; ++j) { sum2[j].x = 0.0f; sum2[j].y = 0.0f; }

    #pragma unroll
    for (int ih = 0; ih < 4; ++ih) {
        #pragma unroll
        for (int iw = 0; iw < 4; ++iw) {
            // Branchless: coefficients are zero whenever the sample is dropped.
            const float c11 = (bx[iw] && by[ih]) ? 0.0f : ax0[iw] * ay0[ih];
            const float c12 = ax0[iw] * ay1[ih];
            const float c21 = ax1[iw] * ay0[ih];
            const float c22 = ax1[iw] * ay1[ih];

            v2f v11[5], v12[5], v21[5], v22[5];
            load10v(&s_ft[ry1[ih] + cx1[iw]], v11);
            load10v(&s_ft[ry2[ih] + cx1[iw]], v12);
            load10v(&s_ft[ry1[ih] + cx2[iw]], v21);
            load10v(&s_ft[ry2[ih] + cx2[iw]], v22);

            // 4 packed FMAs per channel-pair (v_pk_fma_f32), accumulating
            // directly into sum.
            #pragma unroll
            for (int j = 0; j < 5; ++j) {
                sum2[j] = sum2[j] + c11 * v11[j];
                sum2[j] = sum2[j] + c12 * v12[j];
                sum2[j] = sum2[j] + c21 * v21[j];
                sum2[j] = sum2[j] + c22 * v22[j];
            }
        }
    }

    if (n < N) {
        // Single exact division, then multiplies (<=1 ulp vs per-channel div).
        const float invc = (cnt > 0.0f) ? (1.0f / cnt) : 1.0f;
        // out[n, c, bin], layout (N, C, 49); whole output (58.8MB) is L2-resident.
        float* op = out + (size_t)n * (CCH * NBINS) + bin;
        #pragma unroll
        for (int j = 0; j < 5; ++j) {
            op[(2 * j + 0) * NBINS] = sum2[j].x * invc;
            op[(2 * j + 1) * NBINS] = sum2[j].y * invc;
        }
    }
}

extern "C" void kernel_launch(void* const* d_in, const int* in_sizes, int n_in,
                              void* d_out, int out_size, void* d_ws, size_t ws_size,
                              hipStream_t stream) {
    const float* ft   = (const float*)d_in[0];   // (1, C*PH*PW, H, W) f32
    const float* rois = (const float*)d_in[1];   // (N, 5) f32
    float* out = (float*)d_out;                  // (N, C, 49) f32

    const int N = in_sizes[1] / 5;
    dim3 grid((N + ROI_TILE - 1) / ROI_TILE, NBINS);
    dfmb_psroi_align_kernel<<<grid, dim3(ROI_TILE), 0, stream>>>(ft, rois, out, N);
}